// kernel_generated_3_21887153341203
// MI455X (gfx1250) — compile-verified
//
#include <hip/hip_runtime.h>

// CDNA5 / gfx1250, wave32. GEMM formulation of the roll+unfold+einsum:
//   y[i, col] = sum_kk  W2[i, kk] * Bmat[kk, col]
//   M=256 (i), K=384 (kk = l*6 + jj*3 + k, == W row-major), N=196 (col = n*14+o)
// Full-f32 matrix path via V_WMMA_F32_16X16X4_F32 (problem is latency-bound;
// f32 WMMA keeps reference precision at no cost).

typedef __attribute__((ext_vector_type(2))) float v2f;
typedef __attribute__((ext_vector_type(8))) float v8f;

#define K_TOT 384
#define N_COLS 196
#define TILE_N 16

__global__ __launch_bounds__(512) void kernel_roll_unfold_einsum_wmma(
    const float* __restrict__ x,   // (1,2,14,896) f32
    const float* __restrict__ W,   // (256,64,2,3) f32  == (256,384) row-major
    float* __restrict__ y)         // (1,256,14,14) f32 == (256,196)
{
    __shared__ float sB[K_TOT * TILE_N];   // 24 KB B tile (K x 16 cols)

    const int nt  = blockIdx.x;            // N tile: 0..12
    const int tid = threadIdx.x;           // 0..511 (16 waves of 32)

    // ---- Phase 1: build B tile in LDS (im2col gather with rolls folded in) ----
    // sB[kk*16 + c] = Bmat[kk, nt*16 + c]
    for (int e = tid; e < K_TOT * TILE_N; e += 512) {
        const int kk  = e >> 4;
        const int c   = e & 15;
        const int col = nt * TILE_N + c;
        float val = 0.0f;
        if (col < N_COLS) {
            const int l  = kk / 6;
            const int r  = kk - 6 * l;
            const int jj = (r >= 3) ? 1 : 0;
            const int k  = r - 3 * jj;
            const int n  = col / 14;
            const int o  = col - 14 * n;
            const int oo = o + k - 1;                    // unfold, pad (1,1)
            if (oo >= 0 && oo < 14) {
                const int cp = (2 * n + jj + 27) % 28;   // roll along merged 28-axis
                const int hp = cp >> 1;
                const int jp = cp & 1;
                const int so = (oo + 13) % 14;           // roll along last axis
                val = x[jp * 12544 + hp * 896 + l * 14 + so];
            }
        }
        sB[e] = val;
    }
    __syncthreads();

    // ---- Phase 2: one 16x16 output tile per wave via V_WMMA_F32_16X16X4_F32 ----
    const int wave  = tid >> 5;     // 0..15 -> M tile
    const int lane  = tid & 31;
    const int lm    = lane & 15;    // row (A) / col (B,C,D) index
    const int hi    = lane >> 4;    // lane-half selects K pair / M offset
    const int mbase = wave * 16;

    // A frag: lane holds A[m = lm, k = kb + 2*hi + v], v=0,1  -> contiguous float2
    const float* aptr  = W + (size_t)(mbase + lm) * K_TOT + 2 * hi;
    // B frag: lane holds B[k = kb + 2*hi + v, n = lm]
    const float* bbase = sB + lm;

    v8f acc = {0.f, 0.f, 0.f, 0.f, 0.f, 0.f, 0.f, 0.f};

#pragma unroll 4
    for (int kb = 0; kb < K_TOT; kb += 4) {
        v2f a = *(const v2f*)(aptr + kb);                 // global_load_b64
        v2f b;
        b.x = bbase[(kb + 2 * hi)     * TILE_N];          // ds_load_b32
        b.y = bbase[(kb + 2 * hi + 1) * TILE_N];          // ds_load_b32
        // 8-arg form: (neg_a, A, neg_b, B, c_mod, C, reuse_a, reuse_b)
        acc = __builtin_amdgcn_wmma_f32_16x16x4_f32(
            false, a, false, b, (short)0, acc, false, false);
    }

    // C/D layout: VGPR r holds D[m = r + 8*hi, n = lm]
    const int col = nt * TILE_N + lm;
    if (col < N_COLS) {
#pragma unroll
        for (int r = 0; r < 8; ++r) {
            const int i = mbase + r + 8 * hi;
            y[i * N_COLS + col] = acc[r];
        }
    }
}

extern "C" void kernel_launch(void* const* d_in, const int* in_sizes, int n_in,
                              void* d_out, int out_size, void* d_ws, size_t ws_size,
                              hipStream_t stream) {
    const float* x = (const float*)d_in[0];   // (1,2,14,896) f32
    const float* W = (const float*)d_in[1];   // (256,64,2,3) f32
    float* y = (float*)d_out;                 // (1,256,14,14) f32

    // 13 N-tiles (196 cols -> 13x16), 16 waves/block = 16 M-tiles
    kernel_roll_unfold_einsum_wmma<<<dim3(13), dim3(512), 0, stream>>>(x, W, y);
}